// Loss_27479200760039
// MI455X (gfx1250) — compile-verified
//
#include <hip/hip_runtime.h>

#define NCH 30
#define CELLS_PER_BLOCK 256
#define TOTAL_F (CELLS_PER_BLOCK * NCH)   // 7680 floats per tensor per block
#define TOTAL_V4 (TOTAL_F / 4)            // 1920 float4
#define LAMBDA_COORD 5.0f
#define LAMBDA_NOOBJ 0.5f

typedef __attribute__((ext_vector_type(2))) float v2f;
typedef __attribute__((ext_vector_type(8))) float v8f;

// ---------------------------------------------------------------------------
// Kernel 1: per-cell YOLO loss, one partial sum per block.
// Stage 256 cells (2 x 30720 B) into LDS with coalesced float4 loads (14
// unconditional + 2 guarded b128 per thread, issued up front for MLP), then
// each thread computes its cell from LDS (stride-30 word reads: gcd(30,64)=2
// => 32 lanes hit 32 distinct banks, conflict-free), then block tree-reduce.
// ---------------------------------------------------------------------------
__global__ __launch_bounds__(CELLS_PER_BLOCK) void yolo_loss_partial(
    const float* __restrict__ pred, const float* __restrict__ gt,
    float* __restrict__ partials, long long ncells)
{
    __shared__ float lp[TOTAL_F];
    __shared__ float lg[TOTAL_F];

    const int tid = threadIdx.x;
    const long long blockBase = (long long)blockIdx.x * TOTAL_F;

    const float4* __restrict__ pv = (const float4*)(pred + blockBase);
    const float4* __restrict__ gv = (const float4*)(gt + blockBase);
    float4* lpv = (float4*)lp;
    float4* lgv = (float4*)lg;

    const long long remF = ncells * NCH - blockBase;   // floats remaining
    if (remF >= (long long)TOTAL_F) {
        // full tile: 1920 float4 over 256 threads = 7.5 each
        #pragma unroll
        for (int j = 0; j < 7; ++j) {
            const int i = tid + j * CELLS_PER_BLOCK;
            lpv[i] = pv[i];
            lgv[i] = gv[i];
        }
        const int i7 = tid + 7 * CELLS_PER_BLOCK;
        if (i7 < TOTAL_V4) {                // tid < 128
            lpv[i7] = pv[i7];
            lgv[i7] = gv[i7];
        }
    } else {
        const int nv4 = (int)(remF >> 2);
        for (int i = tid; i < nv4; i += CELLS_PER_BLOCK) {
            lpv[i] = pv[i];
            lgv[i] = gv[i];
        }
    }
    __syncthreads();

    const float* p = lp + tid * NCH;
    const float* g = lg + tid * NCH;

    // --- per-cell loss ---
    const float invS = 1.0f / 7.0f;

    const float objf = (g[4] > 0.0f) ? 1.0f : 0.0f;

    // no-object confidence loss (both boxes; gt conf duplicated at 4 and 9)
    float dc0 = p[4] - g[4];
    float dc1 = p[9] - g[9];
    float lnoobj = dc0 * dc0 + dc1 * dc1;

    // gt box in xyxy
    float tw = g[2], th = g[3];
    float tx = g[0] * invS, ty = g[1] * invS;
    float t1x = tx - 0.5f * tw, t1y = ty - 0.5f * th;
    float t2x = tx + 0.5f * tw, t2y = ty + 0.5f * th;
    float ta = tw * th;

    float iou0, iou1;
    #pragma unroll
    for (int b = 0; b < 2; ++b) {
        const int o = 5 * b;
        float pw = p[o + 2], ph = p[o + 3];
        float px = p[o] * invS, py = p[o + 1] * invS;
        float p1x = px - 0.5f * pw, p1y = py - 0.5f * ph;
        float p2x = px + 0.5f * pw, p2y = py + 0.5f * ph;
        float iw = fmaxf(fminf(p2x, t2x) - fmaxf(p1x, t1x), 0.0f);
        float ih = fmaxf(fminf(p2y, t2y) - fmaxf(p1y, t1y), 0.0f);
        float inter = iw * ih;
        float v = inter / (pw * ph + ta - inter);
        if (b == 0) iou0 = v; else iou1 = v;
    }

    // jnp.argmax: first max wins on ties -> box 0 iff iou0 >= iou1
    const int   sel    = (iou0 >= iou1) ? 0 : 5;
    const float maxiou = fmaxf(iou0, iou1);

    float dx = p[sel]     - g[0];
    float dy = p[sel + 1] - g[1];
    float lxy = dx * dx + dy * dy;
    float dw = sqrtf(p[sel + 2]) - sqrtf(g[2]);
    float dh = sqrtf(p[sel + 3]) - sqrtf(g[3]);
    float lwh = dw * dw + dh * dh;
    float dob = p[sel + 4] - maxiou;
    float lobj = dob * dob;

    float lcls = 0.0f;
    #pragma unroll
    for (int k = 10; k < 30; ++k) {
        float d = p[k] - g[k];
        lcls += d * d;
    }

    float cell = objf * (LAMBDA_COORD * (lxy + lwh) + lobj + lcls)
               + (1.0f - objf) * (LAMBDA_NOOBJ * lnoobj);

    const long long cellIdx = (long long)blockIdx.x * CELLS_PER_BLOCK + tid;
    if (cellIdx >= ncells) cell = 0.0f;   // guard (grid divides exactly here)

    // --- block tree reduction, reusing lp ---
    __syncthreads();
    lp[tid] = cell;
    __syncthreads();
    #pragma unroll
    for (int s = CELLS_PER_BLOCK / 2; s > 0; s >>= 1) {
        if (tid < s) lp[tid] += lp[tid + s];
        __syncthreads();
    }
    if (tid == 0) partials[blockIdx.x] = lp[0];
}

// ---------------------------------------------------------------------------
// Kernel 2: 8 waves reduce the (zero-padded to x64) partials array with
// V_WMMA_F32_16X16X4_F32.  With B = all-ones, D = A x B + C accumulates
// sum(all 64 A elems) into every column of C; A's lane layout is a bijection
// onto the 16x4 matrix, so placement of the 64 chunk values is irrelevant.
// Wave w handles chunks w, w+8, ... (branchless global_load_b64 per chunk;
// latency overlapped across the 8 waves).  Column 0 of D lives in lanes 0
// (M=0..7 -> regs 0..7) and 16 (M=8..15): wave total = sum(lane0)+sum(lane16).
// 8 wave totals combine through LDS.
// ---------------------------------------------------------------------------
__global__ __launch_bounds__(256) void final_reduce_wmma(
    const float* __restrict__ partials, float* __restrict__ out,
    int nPadded, float scale)
{
    __shared__ float wsum[8];

    const int tid  = threadIdx.x;
    const int wave = tid >> 5;
    const int lane = tid & 31;

    const float2* __restrict__ pv = (const float2*)partials;   // 8B aligned

    v8f acc = {};
    v2f bones;
    bones.x = 1.0f;
    bones.y = 1.0f;

    const int chunks = nPadded >> 6;              // padded: no guards needed
    for (int c = wave; c < chunks; c += 8) {      // wave-uniform loop bound
        const float2 t = pv[(c << 5) + lane];     // global_load_b64
        v2f a;
        a.x = t.x;
        a.y = t.y;
        // 8 args: (neg_a, A, neg_b, B, c_mod, C, reuse_a, reuse_b)
        acc = __builtin_amdgcn_wmma_f32_16x16x4_f32(
            false, a, false, bones, (short)0, acc, false, false);
    }

    float s = acc[0] + acc[1] + acc[2] + acc[3]
            + acc[4] + acc[5] + acc[6] + acc[7];
    float wtot = __shfl(s, 0, 32) + __shfl(s, 16, 32);
    if (lane == 0) wsum[wave] = wtot;
    __syncthreads();

    if (tid == 0) {
        float total = 0.0f;
        #pragma unroll
        for (int w = 0; w < 8; ++w) total += wsum[w];
        out[0] = total * scale;
    }
}

// ---------------------------------------------------------------------------
extern "C" void kernel_launch(void* const* d_in, const int* in_sizes, int n_in,
                              void* d_out, int out_size, void* d_ws, size_t ws_size,
                              hipStream_t stream) {
    const float* pred = (const float*)d_in[0];
    const float* gt   = (const float*)d_in[1];
    float* out      = (float*)d_out;
    float* partials = (float*)d_ws;

    const long long ncells  = (long long)in_sizes[0] / NCH;                     // 802816
    const int nblocks = (int)((ncells + CELLS_PER_BLOCK - 1) / CELLS_PER_BLOCK); // 3136
    const long long bs = ncells / 49;                                           // 16384

    const int nPadded = (nblocks + 63) & ~63;     // 3136 -> pad = 0 here
    if (nPadded > nblocks) {
        hipMemsetAsync(partials + nblocks, 0,
                       (size_t)(nPadded - nblocks) * sizeof(float), stream);
    }

    yolo_loss_partial<<<nblocks, CELLS_PER_BLOCK, 0, stream>>>(pred, gt, partials, ncells);
    final_reduce_wmma<<<1, 256, 0, stream>>>(partials, out, nPadded, 1.0f / (float)bs);
}